// BatchLsReftIntervention_82952998355115
// MI455X (gfx1250) — compile-verified
//
#include <hip/hip_runtime.h>

#define B_ 8
#define S_ 4096
#define H_ 2048
#define TOPK_ 32

typedef __attribute__((ext_vector_type(2))) float v2f;
typedef __attribute__((ext_vector_type(8))) float v8f;

// ---------------- K1: gather + rank-1 GEMV via V_WMMA_F32_16X16X4_F32 --------
// Block = 256 threads (8 waves). Each block handles 128 gathered rows of one
// batch; each wave owns a 16-row WMMA tile. K walked in chunks of 64 floats,
// double-buffered in LDS, staged with GLOBAL_LOAD_ASYNC_TO_LDS_B128 (ASYNCcnt),
// consumed with conflict-free ds_load_2addr_b64 (row stride 68 -> bank 4r%64).
#define ROWS_ 128
#define KC_ 64
#define LSTR_ 68

__device__ __forceinline__ unsigned int lds_off32(const void* p) {
  // Flat LDS address: low 32 bits are the LDS byte offset (ISA 10.2 aperture).
  return (unsigned int)(unsigned long long)p;
}

__device__ __forceinline__ void async_ld_b128(unsigned int lds_dst, const void* gsrc) {
  asm volatile("global_load_async_to_lds_b128 %0, %1, off"
               :: "v"(lds_dst), "v"((unsigned long long)gsrc) : "memory");
}

__device__ __forceinline__ void wait_async0() {
  asm volatile("s_wait_asynccnt 0x0" ::: "memory");
}

__global__ __launch_bounds__(256) void detect_wmma_kernel(
    const float* __restrict__ base, const int* __restrict__ pos,
    const float* __restrict__ W, float* __restrict__ detect)
{
  __shared__ __align__(16) float lds[2][ROWS_ * LSTR_];
  __shared__ __align__(16) float wlds[H_];
  __shared__ int rowpos[ROWS_];

  const int bid  = blockIdx.x;
  const int b    = bid >> 5;              // S_/ROWS_ == 32 blocks per batch
  const int s0   = (bid & 31) * ROWS_;
  const int tid  = threadIdx.x;
  const int lane = tid & 31;
  const int wave = tid >> 5;

  if (tid < ROWS_) rowpos[tid] = pos[b * S_ + s0 + tid];

  // Stage the whole W row (8 KB) once: 256 threads x 2 async b128.
  {
    const unsigned int wdst = lds_off32(&wlds[tid * 8]);
    const float* wsrc = W + b * H_ + tid * 8;
    async_ld_b128(wdst, wsrc);
    async_ld_b128(wdst + 16, wsrc + 4);
  }
  __syncthreads();                         // rowpos visible for gather addressing

  // A-matrix k mapping (ISA 7.12.2, 32-bit A 16x4): lanes 0-15 -> K+0/K+1,
  // lanes 16-31 -> K+2/K+3. B uses the same k slots, value replicated over N.
  const int m    = lane & 15;
  const int koff = (lane >> 4) << 1;
  const int r    = (tid >> 4);             // staging row within a 16-row pass
  const int c    = (tid & 15) << 2;        // staging column (floats)

  v8f acc = {};

  // Issue chunk 0 into buffer 0.
  #pragma unroll
  for (int p = 0; p < 8; ++p) {
    const int rr = p * 16 + r;
    async_ld_b128(lds_off32(&lds[0][rr * LSTR_ + c]),
                  base + ((size_t)b * S_ + rowpos[rr]) * H_ + c);
  }

  int buf = 0;
  for (int kc = 0; kc < H_; kc += KC_, buf ^= 1) {
    wait_async0();                         // this wave's async writes done
    __syncthreads();                       // => all waves' writes visible

    if (kc + KC_ < H_) {                   // prefetch next chunk into other buf
      #pragma unroll
      for (int p = 0; p < 8; ++p) {
        const int rr = p * 16 + r;
        async_ld_b128(lds_off32(&lds[buf ^ 1][rr * LSTR_ + c]),
                      base + ((size_t)b * S_ + rowpos[rr]) * H_ + kc + KC_ + c);
      }
    }

    const float* arow = &lds[buf][(wave * 16 + m) * LSTR_ + koff];
    const float* wrow = &wlds[kc + koff];  // LDS broadcast reads
    #pragma unroll
    for (int kk = 0; kk < KC_; kk += 4) {
      v2f A;  A.x  = arow[kk]; A.y  = arow[kk + 1];
      v2f Bm; Bm.x = wrow[kk]; Bm.y = wrow[kk + 1];
      acc = __builtin_amdgcn_wmma_f32_16x16x4_f32(
          /*neg_a=*/false, A, /*neg_b=*/false, Bm,
          /*c_mod=*/(short)0, acc, /*reuse_a=*/false, /*reuse_b=*/false);
    }
  }

  // C layout (32-bit 16x16): VGPR v holds M=v (lanes 0-15) and M=v+8 (16-31).
  // Result replicated across N; pull from N==0 of each half.
  const int srow = b * S_ + s0 + wave * 16;
  if (lane == 0) {
    #pragma unroll
    for (int v = 0; v < 8; ++v) detect[srow + v] = fmaxf(acc[v], 0.0f);
  } else if (lane == 16) {
    #pragma unroll
    for (int v = 0; v < 8; ++v) detect[srow + 8 + v] = fmaxf(acc[v], 0.0f);
  }
}

// ---------------- K2: per-batch top-32, non_topk, mean, position histogram ---
__global__ __launch_bounds__(256) void topk_kernel(
    const float* __restrict__ detect, const int* __restrict__ pos,
    float* __restrict__ nontopk, int* __restrict__ counts,
    float* __restrict__ scale)
{
  __shared__ float vals[S_];
  __shared__ float rv[256];
  __shared__ int   ri[256];

  const int b   = blockIdx.x;
  const int tid = threadIdx.x;

  for (int i = tid; i < S_; i += 256) {
    vals[i] = detect[b * S_ + i];
    counts[b * S_ + i] = 0;               // re-init workspace every call
  }
  __syncthreads();
  __threadfence();

  float topsum = 0.0f;
  for (int it = 0; it < TOPK_; ++it) {
    float bv = -1.0f; int bi = 0x7fffffff;
    for (int i = tid; i < S_; i += 256) {
      const float v = vals[i];
      if (v > bv || (v == bv && i < bi)) { bv = v; bi = i; }
    }
    rv[tid] = bv; ri[tid] = bi;
    __syncthreads();
    for (int off = 128; off > 0; off >>= 1) {
      if (tid < off) {
        if (rv[tid + off] > rv[tid] ||
            (rv[tid + off] == rv[tid] && ri[tid + off] < ri[tid])) {
          rv[tid] = rv[tid + off]; ri[tid] = ri[tid + off];
        }
      }
      __syncthreads();
    }
    if (tid == 0) { topsum += rv[0]; vals[ri[0]] = -1.0f; }  // sentinel (relu>=0)
    __syncthreads();
  }
  if (tid == 0) scale[b] = topsum * (1.0f / (float)TOPK_);

  for (int i = tid; i < S_; i += 256) {
    const float v = vals[i];
    nontopk[b * S_ + i] = (v < 0.0f) ? 0.0f : v;             // topk zeroed
    atomicAdd(&counts[b * S_ + pos[b * S_ + i]], 1);         // scatter histogram
  }
}

// ---------------- K3: mixed = base + count[b,p] * mean * W  (streaming) ------
__global__ __launch_bounds__(512) void mix_kernel(
    const float* __restrict__ base, const float* __restrict__ W,
    const int* __restrict__ counts, const float* __restrict__ scale,
    float* __restrict__ out)
{
  const int row  = blockIdx.x;            // b*S_ + p
  const int b    = row >> 12;
  const int tid  = threadIdx.x;
  const float cf = scale[b] * (float)counts[row];
  const size_t off = (size_t)row * H_ + (size_t)tid * 4;

  float4 v = *(const float4*)(base + off);
  const float4 w = *(const float4*)(W + b * H_ + tid * 4);
  v.x += cf * w.x; v.y += cf * w.y; v.z += cf * w.z; v.w += cf * w.w;
  *(float4*)(out + off) = v;
}

// -----------------------------------------------------------------------------
extern "C" void kernel_launch(void* const* d_in, const int* in_sizes, int n_in,
                              void* d_out, int out_size, void* d_ws, size_t ws_size,
                              hipStream_t stream) {
  const float* base = (const float*)d_in[0];
  const int*   pos  = (const int*)d_in[1];
  const float* W    = (const float*)d_in[2];

  float* out_mixed   = (float*)d_out;
  float* out_detect  = out_mixed + (size_t)B_ * S_ * H_;
  float* out_nontopk = out_detect + (size_t)B_ * S_;

  int*   counts = (int*)d_ws;                                     // B*S ints
  float* scale  = (float*)((char*)d_ws + (size_t)B_ * S_ * sizeof(int)); // B floats

  detect_wmma_kernel<<<(B_ * S_) / ROWS_, 256, 0, stream>>>(base, pos, W, out_detect);
  topk_kernel<<<B_, 256, 0, stream>>>(out_detect, pos, out_nontopk, counts, scale);
  mix_kernel<<<B_ * S_, 512, 0, stream>>>(base, W, counts, scale, out_mixed);
}